// MLA_29008209117518
// MI455X (gfx1250) — compile-verified
//
#include <hip/hip_runtime.h>
#include <hip/hip_bf16.h>
#include <math.h>

typedef __attribute__((ext_vector_type(2))) float v2f;
typedef __attribute__((ext_vector_type(8))) float v8f;
typedef __attribute__((address_space(3))) float lds_f;

static __device__ __forceinline__ v8f wmma_f32(v2f a, v2f b, v8f c) {
  // D = A(16x4 f32) x B(4x16 f32) + C(16x16 f32)
  return __builtin_amdgcn_wmma_f32_16x16x4_f32(
      /*neg_a=*/false, a, /*neg_b=*/false, b,
      /*c_mod=*/(short)0, c, /*reuse_a=*/false, /*reuse_b=*/false);
}

// Async global->LDS copy (ASYNCcnt path, no VGPR round trip).
// VDST = 32-bit LDS byte offset, VADDR = 64-bit global address.
static __device__ __forceinline__ void async_b128(const float* gsrc, float* ldst) {
  asm volatile("global_load_async_to_lds_b128 %0, %1, off"
               :: "v"((lds_f*)ldst), "v"(gsrc) : "memory");
}
static __device__ __forceinline__ void wait_async_0()  { asm volatile("s_wait_asynccnt 0x0" ::: "memory"); }
static __device__ __forceinline__ void wait_async_4()  { asm volatile("s_wait_asynccnt 0x4" ::: "memory"); }
static __device__ __forceinline__ void wait_async_10() { asm volatile("s_wait_asynccnt 0xa" ::: "memory"); }

// ---------------------------------------------------------------------------
// fp32 GEMM + bias: C[M,N] = A[M,K] @ W[K,N] + bias[N]
// M%64==0, N%64==0, K%16==0 (true at every call site).
// 128 threads = 4 waves, each computing a 32x32 patch (2x2 WMMA tiles).
// Double-buffered async global->LDS staging (4 b128 ops/thread/stage).
// ---------------------------------------------------------------------------
#define BM 64
#define BN 64
#define BK 16
#define ASTR 20   // A row stride (floats): 16B-aligned rows, 20r mod 64 distinct

__global__ __launch_bounds__(128)
void gemm_bias_f32(const float* __restrict__ A, const float* __restrict__ W,
                   const float* __restrict__ bias, float* __restrict__ C,
                   int M, int N, int K) {
  __shared__ float As[2][BM][ASTR];
  __shared__ float Bs[2][BK][BN];

  const int tid    = threadIdx.x;
  const int wave   = tid >> 5;
  const int lane   = tid & 31;
  const int lm     = lane & 15;
  const int khalf  = (lane >> 4) * 2;   // K sub-offset: lanes 0-15 -> 0, 16-31 -> 2
  const int rowadd = (lane >> 4) * 8;   // C layout: rows r / r+8 across lane halves

  const int m0 = blockIdx.y * BM;
  const int n0 = blockIdx.x * BN;
  const int wm = (wave >> 1) * 32;      // wave grid 2(m) x 2(n)
  const int wn = (wave & 1) * 32;

  v8f acc[2][2];
  for (int i = 0; i < 2; ++i)
    for (int j = 0; j < 2; ++j)
      for (int r = 0; r < 8; ++r) acc[i][j][r] = 0.0f;

  auto load_stage = [&](int k0, int buf) {
    // A tile 64x16: 256 b128 segs, 2/thread
    for (int e = 0; e < 2; ++e) {
      const int idx = tid + e * 128;
      const int r = idx >> 2, c = (idx & 3) * 4;
      async_b128(A + (size_t)(m0 + r) * K + (k0 + c), &As[buf][r][c]);
    }
    // B tile 16x64: 256 b128 segs, 2/thread
    for (int e = 0; e < 2; ++e) {
      const int idx = tid + e * 128;
      const int r = idx >> 4, c = (idx & 15) * 4;
      async_b128(W + (size_t)(k0 + r) * N + (n0 + c), &Bs[buf][r][c]);
    }
  };

  const int NK = K / BK;
  load_stage(0, 0);

  for (int kb = 0; kb < NK; ++kb) {
    const int buf = kb & 1;
    if (kb + 1 < NK) {
      load_stage((kb + 1) * BK, buf ^ 1);
      wait_async_4();                 // stage kb landed; stage kb+1 in flight
    } else {
      wait_async_0();
    }
    __syncthreads();

    for (int ks = 0; ks < 4; ++ks) {
      const int kc = ks * 4 + khalf;
      v2f a[2], b[2];
      for (int i = 0; i < 2; ++i) {
        a[i].x = As[buf][wm + i * 16 + lm][kc];
        a[i].y = As[buf][wm + i * 16 + lm][kc + 1];
      }
      for (int j = 0; j < 2; ++j) {
        b[j].x = Bs[buf][kc][wn + j * 16 + lm];
        b[j].y = Bs[buf][kc + 1][wn + j * 16 + lm];
      }
      for (int i = 0; i < 2; ++i)
        for (int j = 0; j < 2; ++j)
          acc[i][j] = wmma_f32(a[i], b[j], acc[i][j]);
    }
    __syncthreads();                  // publish before buf^1 gets overwritten
  }

  for (int i = 0; i < 2; ++i)
    for (int j = 0; j < 2; ++j) {
      const int n = n0 + wn + j * 16 + lm;
      const float bv = bias[n];
      for (int r = 0; r < 8; ++r) {
        const int m = m0 + wm + i * 16 + r + rowadd;
        C[(size_t)m * N + n] = acc[i][j][r] + bv;
      }
    }
}

// ---------------------------------------------------------------------------
// RoPE (faithful to reference: cos_pos holds sin, sin_pos holds cos):
//   out[2i]   = q[2i]*sin(p*th) - q[2i+1]*cos(p*th)
//   out[2i+1] = q[2i+1]*sin(p*th) + q[2i]*cos(p*th)
//   th_i = 10000^(-(2i)/64) = exp(i * (-ln(10000)/32))
// ---------------------------------------------------------------------------
#define ROPE_LN_COEF (-0.28782313662398193f)

__global__ __launch_bounds__(256)
void rope_qr_kernel(float* __restrict__ qr) {
  const int gid = blockIdx.x * 256 + threadIdx.x;   // B*S*H*32 = 4194304
  const int row = gid >> 9;
  const int rem = gid & 511;
  const int hh  = rem >> 5;
  const int i   = rem & 31;
  const int s   = row & 511;                        // S = 512
  const float ang = (float)s * expf((float)i * ROPE_LN_COEF);
  const float sn = sinf(ang), cs = cosf(ang);
  float* p = qr + (size_t)row * 1024 + hh * 64 + 2 * i;
  const float q0 = p[0], q1 = p[1];
  p[0] = q0 * sn - q1 * cs;
  p[1] = q1 * sn + q0 * cs;
}

__global__ __launch_bounds__(256)
void rope_kr_kernel(float* __restrict__ kr) {
  const int gid = blockIdx.x * 256 + threadIdx.x;   // B*S*32 = 262144
  const int row = gid >> 5;
  const int i   = gid & 31;
  const int s   = row & 511;
  const float ang = (float)s * expf((float)i * ROPE_LN_COEF);
  const float sn = sinf(ang), cs = cosf(ang);
  float* p = kr + (size_t)row * 64 + 2 * i;
  const float q0 = p[0], q1 = p[1];
  p[0] = q0 * sn - q1 * cs;
  p[1] = q1 * sn + q0 * cs;
}

// ---------------------------------------------------------------------------
// Flash attention: one block per (b, h, 64-row q tile); 4 waves, wave w owns
// q rows [w*16, w*16+16) so the key-direction softmax is wave-private.
// d = 192 = [q_c(128) | roped q_r(64)]; v head dim = 128.
// K/V tiles double-buffered via async global->LDS (10 b128 ops/thread/stage).
// ---------------------------------------------------------------------------
#define QSTR 196   // 16B-aligned rows, 4r mod 64 distinct -> conflict-free cols

__global__ __launch_bounds__(128)
void mla_attention(const float* __restrict__ qc, const float* __restrict__ qr,
                   const float* __restrict__ kc, const float* __restrict__ kr,
                   const float* __restrict__ vc, const int* __restrict__ mask,
                   float* __restrict__ ctx) {
  __shared__ float Qs[64][QSTR];
  __shared__ float Ks[2][16][QSTR];
  __shared__ float Vs[2][16][128];
  __shared__ float Ps[4][16][17];

  const int tid    = threadIdx.x;
  const int wave   = tid >> 5;
  const int lane   = tid & 31;
  const int lm     = lane & 15;
  const int khalf  = (lane >> 4) * 2;
  const int rowadd = (lane >> 4) * 8;

  const int qt = blockIdx.x;           // 0..7
  const int h  = blockIdx.y;           // 0..15
  const int b  = blockIdx.z;           // 0..15
  const size_t rowbase = (size_t)b * 512;
  const float qscale = 1.0f / 19.31370849898476f;  // 1/(sqrt(128)+sqrt(64))

  auto load_kv = [&](int sk0, int buf) {
    // K tile 16x192: 768 b128 segs, 6/thread
    for (int e = 0; e < 6; ++e) {
      const int idx = tid + e * 128;
      const int r = idx / 48, c = (idx % 48) * 4;
      const size_t grow = rowbase + sk0 + r;
      const float* src = (c < 128) ? (kc + grow * 2048 + h * 128 + c)
                                   : (kr + grow * 64 + (c - 128));
      async_b128(src, &Ks[buf][r][c]);
    }
    // V tile 16x128: 512 b128 segs, 4/thread
    for (int e = 0; e < 4; ++e) {
      const int idx = tid + e * 128;
      const int r = idx >> 5, c = (idx & 31) * 4;
      const size_t grow = rowbase + sk0 + r;
      async_b128(vc + grow * 2048 + h * 128 + c, &Vs[buf][r][c]);
    }
  };

  // Q tile 64x192 (raw; score scale applied post-WMMA): 3072 segs, 24/thread
  for (int e = 0; e < 24; ++e) {
    const int idx = tid + e * 128;
    const int r = idx / 48, c = (idx % 48) * 4;
    const size_t grow = rowbase + qt * 64 + r;
    const float* src = (c < 128) ? (qc + grow * 2048 + h * 128 + c)
                                 : (qr + grow * 1024 + h * 64 + (c - 128));
    async_b128(src, &Qs[r][c]);
  }
  load_kv(0, 0);
  wait_async_0();
  __syncthreads();

  // Preload Q A-fragments for all 48 k-steps (d = 192, WMMA K-step = 4)
  v2f qa[48];
  for (int ks = 0; ks < 48; ++ks) {
    const int kc0 = ks * 4 + khalf;
    qa[ks].x = Qs[wave * 16 + lm][kc0];
    qa[ks].y = Qs[wave * 16 + lm][kc0 + 1];
  }

  v8f acc[8];
  for (int c = 0; c < 8; ++c)
    for (int r = 0; r < 8; ++r) acc[c][r] = 0.0f;
  float mstate[8], lstate[8];
  for (int r = 0; r < 8; ++r) { mstate[r] = -1e30f; lstate[r] = 0.0f; }

  for (int kt = 0; kt < 32; ++kt) {
    const int buf = kt & 1;
    const int sk0 = kt * 16;
    if (kt + 1 < 32) {
      load_kv((kt + 1) * 16, buf ^ 1);
      wait_async_10();                // stage kt landed; stage kt+1 in flight
    } else {
      wait_async_0();
    }
    __syncthreads();

    // S_tile(16q x 16k) = Q @ K^T  (B[k][n] = Ks[key=n][d=k])
    v8f sc;
    for (int r = 0; r < 8; ++r) sc[r] = 0.0f;
    for (int ks = 0; ks < 48; ++ks) {
      const int kc0 = ks * 4 + khalf;
      v2f kb;
      kb.x = Ks[buf][lm][kc0];
      kb.y = Ks[buf][lm][kc0 + 1];
      sc = wmma_f32(qa[ks], kb, sc);
    }
    for (int r = 0; r < 8; ++r) sc[r] *= qscale;

    // Key mask (column = key sk0+lm)
    if (mask[rowbase + sk0 + lm] == 0)
      for (int r = 0; r < 8; ++r) sc[r] = -1e30f;

    // Online softmax; row r (+8 upper half) lives in a 16-lane half per VGPR
    float pr[8];
    for (int r = 0; r < 8; ++r) {
      float tmax = sc[r];
      for (int off = 8; off >= 1; off >>= 1)
        tmax = fmaxf(tmax, __shfl_xor(tmax, off, 32));
      const float mnew = fmaxf(mstate[r], tmax);
      const float corr = expf(mstate[r] - mnew);
      const float p = expf(sc[r] - mnew);
      float psum = p;
      for (int off = 8; off >= 1; off >>= 1)
        psum += __shfl_xor(psum, off, 32);
      lstate[r] = lstate[r] * corr + psum;
      mstate[r] = mnew;
      pr[r] = p;
      for (int c = 0; c < 8; ++c) acc[c][r] *= corr;
    }

    // Transpose P (C-layout -> A-layout) through wave-private LDS patch
    for (int r = 0; r < 8; ++r)
      Ps[wave][r + rowadd][lm] = pr[r];
    asm volatile("s_wait_dscnt 0x0" ::: "memory");  // within-wave LDS RAW

    v2f pa[4];
    for (int ks = 0; ks < 4; ++ks) {
      const int kc0 = ks * 4 + khalf;
      pa[ks].x = Ps[wave][lm][kc0];
      pa[ks].y = Ps[wave][lm][kc0 + 1];
    }
    // acc += P(16x16) @ V(16x128), 8 column tiles
    for (int c = 0; c < 8; ++c) {
      for (int ks = 0; ks < 4; ++ks) {
        const int kc0 = ks * 4 + khalf;
        v2f vb;
        vb.x = Vs[buf][kc0][c * 16 + lm];
        vb.y = Vs[buf][kc0 + 1][c * 16 + lm];
        acc[c] = wmma_f32(pa[ks], vb, acc[c]);
      }
    }
    __syncthreads();                  // publish before buf^1 gets overwritten
  }

  // Normalize and write ctx in [B,S,H*128] order
  for (int r = 0; r < 8; ++r) {
    const float inv = 1.0f / lstate[r];
    const int m = qt * 64 + wave * 16 + r + rowadd;
    const size_t grow = rowbase + m;
    for (int c = 0; c < 8; ++c)
      ctx[grow * 2048 + h * 128 + c * 16 + lm] = acc[c][r] * inv;
  }
}

// ---------------------------------------------------------------------------
// Host-side launcher
// ---------------------------------------------------------------------------
extern "C" void kernel_launch(void* const* d_in, const int* in_sizes, int n_in,
                              void* d_out, int out_size, void* d_ws, size_t ws_size,
                              hipStream_t stream) {
  (void)in_sizes; (void)n_in; (void)out_size; (void)ws_size;
  const float* h     = (const float*)d_in[0];
  const int*   mask  = (const int*)  d_in[1];
  const float* W_dkv = (const float*)d_in[2];
  const float* b_dkv = (const float*)d_in[3];
  const float* W_dq  = (const float*)d_in[4];
  const float* b_dq  = (const float*)d_in[5];
  const float* W_uk  = (const float*)d_in[6];
  const float* b_uk  = (const float*)d_in[7];
  const float* W_uv  = (const float*)d_in[8];
  const float* b_uv  = (const float*)d_in[9];
  const float* W_uq  = (const float*)d_in[10];
  const float* b_uq  = (const float*)d_in[11];
  const float* W_qr  = (const float*)d_in[12];
  const float* b_qr  = (const float*)d_in[13];
  const float* W_kr  = (const float*)d_in[14];
  const float* b_kr  = (const float*)d_in[15];
  const float* W_fc  = (const float*)d_in[16];
  const float* b_fc  = (const float*)d_in[17];

  const int M = 16 * 512;  // 8192

  float* ws   = (float*)d_ws;
  float* c_kv = ws;                      // [8192, 512]
  float* c_q  = c_kv + (size_t)4194304;  // [8192, 512]
  float* kc_  = c_q  + (size_t)4194304;  // [8192, 2048] heads interleaved
  float* vc_  = kc_  + (size_t)16777216;
  float* qc_  = vc_  + (size_t)16777216;
  float* qr_  = qc_  + (size_t)16777216; // [8192, 1024]
  float* kr_  = qr_  + (size_t)8388608;  // [8192, 64]
  float* ctx_ = kr_  + (size_t)524288;   // [8192, 2048]

  const dim3 blk(128);
  #define GRID(MM, NN) dim3((NN) / 64, (MM) / 64)

  gemm_bias_f32<<<GRID(M, 512),  blk, 0, stream>>>(h,    W_dkv, b_dkv, c_kv, M, 512,  2048);
  gemm_bias_f32<<<GRID(M, 512),  blk, 0, stream>>>(h,    W_dq,  b_dq,  c_q,  M, 512,  2048);
  gemm_bias_f32<<<GRID(M, 2048), blk, 0, stream>>>(c_kv, W_uk,  b_uk,  kc_,  M, 2048, 512);
  gemm_bias_f32<<<GRID(M, 2048), blk, 0, stream>>>(c_kv, W_uv,  b_uv,  vc_,  M, 2048, 512);
  gemm_bias_f32<<<GRID(M, 2048), blk, 0, stream>>>(c_q,  W_uq,  b_uq,  qc_,  M, 2048, 512);
  gemm_bias_f32<<<GRID(M, 1024), blk, 0, stream>>>(c_q,  W_qr,  b_qr,  qr_,  M, 1024, 512);
  gemm_bias_f32<<<GRID(M, 64),   blk, 0, stream>>>(h,    W_kr,  b_kr,  kr_,  M, 64,   2048);

  rope_qr_kernel<<<(8192 * 512) / 256, 256, 0, stream>>>(qr_);
  rope_kr_kernel<<<(8192 * 32)  / 256, 256, 0, stream>>>(kr_);

  mla_attention<<<dim3(8, 16, 16), blk, 0, stream>>>(qc_, qr_, kc_, kr_, vc_, mask, ctx_);

  gemm_bias_f32<<<GRID(M, 2048), blk, 0, stream>>>(ctx_, W_fc, b_fc, (float*)d_out, M, 2048, 2048);
  #undef GRID
}